// MultiShapeNet_90048284327989
// MI455X (gfx1250) — compile-verified
//
#include <hip/hip_runtime.h>

#define B_  4
#define C_  3
#define N_  8192
#define O_  64
#define K_  20

typedef __attribute__((ext_vector_type(2))) float v2f;
typedef __attribute__((ext_vector_type(8))) float v8f;

// ---------------------------------------------------------------------------
// Kernel 0: pack points as (x, y, z, -|x|^2/2) for WMMA operand loads.
// One b64 load then covers either (x,y) [K=0,1] or (z,w) [K=2,3] per lane.
// ---------------------------------------------------------------------------
__global__ void prep_kernel(const float* __restrict__ x, float* __restrict__ xt4) {
    int i = blockIdx.x * blockDim.x + threadIdx.x;          // over B_*N_
    if (i >= B_ * N_) return;
    int b = i / N_, n = i - b * N_;
    const float* xb = x + (size_t)b * C_ * N_;
    float xv = xb[n];
    float yv = xb[N_ + n];
    float zv = xb[2 * N_ + n];
    float w  = -0.5f * (xv * xv + yv * yv + zv * zv);
    float4 o = make_float4(xv, yv, zv, w);
    *(float4*)(xt4 + (size_t)i * 4) = o;
}

// ---------------------------------------------------------------------------
// Kernel 1: KNN via V_WMMA_F32_16X16X4_F32 Gram tiles.
// Wave handles 32 queries (two 16x16x4 A tiles). D[q][c] = dot3 - |c|^2/2 is
// monotonic in pd for fixed q, so top-k compares D directly.
//   A layout (32b 16x4):  lane l, vgpr v -> M = l%16, K = 2*(l/16)+v
//   B layout (32b 4x16):  lane l, vgpr v -> N = l%16, K = 2*(l/16)+v
//   D layout (32b 16x16): lane l, vgpr r -> M = r+8*(l/16), N = l%16
// LDS transpose slice is private per wave: per-wave in-order DS pipeline +
// a compiler-only wave_barrier replaces workgroup barriers. B operand is
// software-pipelined two tiles deep (branchless via clamped index).
// ---------------------------------------------------------------------------
__global__ __launch_bounds__(128) void knn_kernel(const float* __restrict__ xt4,
                                                  int* __restrict__ gidx) {
    const int lane = threadIdx.x & 31;
    const int wave = threadIdx.x >> 5;
    const int b    = blockIdx.y;
    const int qbase = blockIdx.x * 128 + wave * 32;
    const int m16  = lane & 15;
    const int half = lane >> 4;
    const int k0   = half * 2;

    __shared__ float ldsT[4][16][33];   // [wave][candidate][query col], padded

    const float* xb = xt4 + (size_t)b * N_ * 4;

    // A operands: two query tiles; force A[q][3] = 1.0 so D folds -|c|^2/2.
    v2f a0, a1;
    {
        v2f av = *(const v2f*)(xb + (qbase + m16) * 4 + k0);
        if (half) av.y = 1.0f;
        a0 = av;
        av = *(const v2f*)(xb + (qbase + 16 + m16) * 4 + k0);
        if (half) av.y = 1.0f;
        a1 = av;
    }

    float vals[K_]; int idxs[K_];
#pragma unroll
    for (int i = 0; i < K_; ++i) { vals[i] = -3.0e38f; idxs[i] = 0; }
    float mv = -3.0e38f; int mp = 0;

    // Depth-2 software pipeline on the B operand (candidate tile coords).
    v2f bvA = *(const v2f*)(xb + (0  + m16) * 4 + k0);
    v2f bvB = *(const v2f*)(xb + (16 + m16) * 4 + k0);

    for (int m0 = 0; m0 < N_; m0 += 16) {
        v2f cur = bvA;
        bvA = bvB;
        int nm = (m0 + 32 < N_) ? (m0 + 32) : 0;   // branchless prefetch
        bvB = *(const v2f*)(xb + (nm + m16) * 4 + k0);

        v8f acc0 = {}, acc1 = {};
        acc0 = __builtin_amdgcn_wmma_f32_16x16x4_f32(false, a0, false, cur,
                                                     (short)0, acc0, false, false);
        acc1 = __builtin_amdgcn_wmma_f32_16x16x4_f32(false, a1, false, cur,
                                                     (short)0, acc1, false, false);

        // Transpose through LDS: row = candidate (this lane's N), col = wave-query.
        float* d0 = &ldsT[wave][m16][8 * half];
        float* d1 = &ldsT[wave][m16][16 + 8 * half];
#pragma unroll
        for (int r = 0; r < 8; ++r) { d0[r] = acc0[r]; d1[r] = acc1[r]; }

        // Wave-private LDS slice: DS ops from one wave complete in order, so a
        // compile-time ordering fence is enough (no workgroup barrier needed).
        __builtin_amdgcn_wave_barrier();

        // Gather this lane's 16 candidate values up-front (pairs -> ds_load_2addr).
        float dv[16];
#pragma unroll
        for (int j = 0; j < 16; ++j) dv[j] = ldsT[wave][j][lane];

#pragma unroll
        for (int j = 0; j < 16; ++j) {
            float v = dv[j];
            int id = m0 + j;
            if (v > mv) {                       // rare: ~k*ln(N/k) hits total
#pragma unroll
                for (int i = 0; i < K_; ++i) {  // static-index select: stays in VGPRs
                    bool rep = (i == mp);
                    vals[i] = rep ? v  : vals[i];
                    idxs[i] = rep ? id : idxs[i];
                }
                mv = vals[0]; mp = 0;
#pragma unroll
                for (int i = 1; i < K_; ++i)
                    if (vals[i] < mv) { mv = vals[i]; mp = i; }
            }
        }
        __builtin_amdgcn_wave_barrier();   // keep next iter's stores after reads
    }

    int q = qbase + lane;
    int* op = gidx + ((size_t)b * N_ + q) * K_;
#pragma unroll
    for (int i = 0; i < K_; ++i) op[i] = idxs[i];
}

// ---------------------------------------------------------------------------
// Kernel 2: EdgeConv: feat = [nb - cen, cen] (6) x W(64x6) + b, lrelu(0.2),
// max over k. Thread = (point, chunk of 16 outputs); W/bias staged in LDS.
// ---------------------------------------------------------------------------
__global__ __launch_bounds__(128) void edge_kernel(const float* __restrict__ xt4,
                                                   const int* __restrict__ gidx,
                                                   const float* __restrict__ W,
                                                   const float* __restrict__ bias,
                                                   float* __restrict__ out) {
    __shared__ float Wl[O_ * 6];
    __shared__ float bl[O_];
    for (int i = threadIdx.x; i < O_ * 6; i += 128) Wl[i] = W[i];
    if (threadIdx.x < O_) bl[threadIdx.x] = bias[threadIdx.x];
    __syncthreads();

    const int oc = threadIdx.x >> 5;            // output chunk 0..3 (16 each)
    const int pt = threadIdx.x & 31;
    const int b  = blockIdx.y;
    const int n  = blockIdx.x * 32 + pt;

    const float* xb = xt4 + (size_t)b * N_ * 4;
    const float4 cen4 = *(const float4*)(xb + (size_t)n * 4);
    const float cx = cen4.x, cy = cen4.y, cz = cen4.z;

    const int* ip = gidx + ((size_t)b * N_ + n) * K_;

    float acc[16];
#pragma unroll
    for (int o = 0; o < 16; ++o) acc[o] = -3.0e38f;

    for (int kk = 0; kk < K_; ++kk) {
        int nid = ip[kk];
        const float4 nb4 = *(const float4*)(xb + (size_t)nid * 4);
        float f0 = nb4.x - cx, f1 = nb4.y - cy, f2 = nb4.z - cz;

#pragma unroll
        for (int o = 0; o < 16; ++o) {
            const float* w = &Wl[(oc * 16 + o) * 6];
            float v = bl[oc * 16 + o];
            v += w[0] * f0 + w[1] * f1 + w[2] * f2;
            v += w[3] * cx + w[4] * cy + w[5] * cz;
            v = (v > 0.0f) ? v : 0.2f * v;              // leaky_relu(0.2)
            acc[o] = fmaxf(acc[o], v);
        }
    }

#pragma unroll
    for (int o = 0; o < 16; ++o)
        out[((size_t)b * O_ + (oc * 16 + o)) * N_ + n] = acc[o];
}

// ---------------------------------------------------------------------------
extern "C" void kernel_launch(void* const* d_in, const int* in_sizes, int n_in,
                              void* d_out, int out_size, void* d_ws, size_t ws_size,
                              hipStream_t stream) {
    const float* x    = (const float*)d_in[0];   // (4,3,8192)
    const float* W    = (const float*)d_in[1];   // (64,6)
    const float* bias = (const float*)d_in[2];   // (64,)
    // d_in[3] = k (== 20, hardcoded)

    float* xt4 = (float*)d_ws;                                   // B*N*4 floats
    int*   gidx = (int*)((char*)d_ws + (size_t)B_ * N_ * 4 * sizeof(float)); // B*N*20 ints
    float* out = (float*)d_out;                                  // (4,64,8192)

    prep_kernel<<<(B_ * N_ + 255) / 256, 256, 0, stream>>>(x, xt4);
    knn_kernel<<<dim3(N_ / 128, B_), 128, 0, stream>>>(xt4, gidx);
    edge_kernel<<<dim3(N_ / 32, B_), 128, 0, stream>>>(xt4, gidx, W, bias, out);
}